// NNGPKernel_30554397344001
// MI455X (gfx1250) — compile-verified
//
#include <hip/hip_runtime.h>

// NNGP initial ConvKP tensors on MI455X (gfx1250, wave32, WMMA).
// xy[n1*N2+n2, wh] = (1/C) * sum_c x[n1,c,wh] * y[n2,c,wh]   (per-wh 128x128x768 GEMM)
// xx[n1, wh]       = (1/C) * sum_c x[n1,c,wh]^2              (fused into staging)
// yy[n2, wh]       = (1/C) * sum_c y[n2,c,wh]^2
//
// LDS tiles are stored K-paired as float2 so each WMMA A/B fragment is a single
// aligned ds_load_b64 directly into the operand VGPR pair (no re-pairing movs).
// Double-buffered K pipeline: one barrier per tile, global loads for tile t+1
// overlap the 16 WMMAs of tile t.

typedef float v2f __attribute__((ext_vector_type(2)));
typedef float v8f __attribute__((ext_vector_type(8)));

constexpr int N1_   = 128;
constexpr int N2_   = 128;
constexpr int C_DIM = 768;
constexpr int WH    = 256;        // 16*16 spatial positions
constexpr int KT    = 16;         // K tile per pipeline stage
constexpr int KP    = KT / 2;     // K-pairs per tile
constexpr int LDN   = 144;        // float2 row stride: 2*144 % 64 == 32 -> lane halves on disjoint banks

__global__ __launch_bounds__(512)
void nngp_convkp_kernel(const float* __restrict__ x,
                        const float* __restrict__ y,
                        float* __restrict__ out_xy,
                        float* __restrict__ out_xx,
                        float* __restrict__ out_yy)
{
    __shared__ v2f  XsP[2][KP * LDN];   // [buf][kpair][n1] as {X[2kp][n], X[2kp+1][n]}
    __shared__ v2f  YsP[2][KP * LDN];   // [buf][kpair][n2]
    __shared__ float red[8 * 128];      // 4x128 partials for xx, 4x128 for yy

    const int wh   = blockIdx.x;        // one spatial position per block
    const int t    = threadIdx.x;       // 0..511 (16 wave32s)
    const int lane = t & 31;
    const int wave = t >> 5;

    // 4x4 grid of 32x32 wave tiles over the 128x128 output
    const int m0  = (wave >> 2) * 32;
    const int n0  = (wave & 3) * 32;
    const int row = lane & 15;          // M (A) / N (B) within the 16x16 tile
    const int kh  = lane >> 4;          // lane half selects K-pair +0 / +1

    const size_t nstride = (size_t)C_DIM * WH;

    v8f acc[2][2] = {};
    float sxx = 0.f, syy = 0.f;

    // Staging map: element e = t + 512*j -> n = t&127 (fixed), c = (t>>7) + 4*j
    const int    nloc  = t & 127;
    const int    cbase = t >> 7;
    const size_t gbase = (size_t)nloc * nstride + (size_t)wh;

    float xr[4], yr[4];

    // ---- prologue: load + stage tile 0 ----
    #pragma unroll
    for (int j = 0; j < 4; ++j) {
        const int c = cbase + 4 * j;
        xr[j] = x[gbase + (size_t)c * WH];
        yr[j] = y[gbase + (size_t)c * WH];
        sxx += xr[j] * xr[j];
        syy += yr[j] * yr[j];
    }
    {
        float* xb = (float*)&XsP[0][0];
        float* yb = (float*)&YsP[0][0];
        #pragma unroll
        for (int j = 0; j < 4; ++j) {
            const int c = cbase + 4 * j;
            const int idx = ((c >> 1) * LDN + nloc) * 2 + (c & 1);
            xb[idx] = xr[j];
            yb[idx] = yr[j];
        }
    }
    __syncthreads();

    const int NT = C_DIM / KT;          // 48 pipeline stages
    for (int kt = 0; kt < NT; ++kt) {
        const int cur = kt & 1;

        // ---- issue global loads for tile kt+1 (overlap with WMMA below) ----
        const bool have_next = (kt + 1) < NT;
        if (have_next) {
            const int k0n = (kt + 1) * KT;
            #pragma unroll
            for (int j = 0; j < 4; ++j) {
                const int c = k0n + cbase + 4 * j;
                xr[j] = x[gbase + (size_t)c * WH];
                yr[j] = y[gbase + (size_t)c * WH];
            }
        }

        // ---- 16 WMMAs on the current tile ----
        const v2f* Xp = &XsP[cur][0];
        const v2f* Yp = &YsP[cur][0];
        #pragma unroll
        for (int kk = 0; kk < KP; kk += 2) {          // 2 K-pairs (=4 K) per step
            const int kpl = kk + kh;                  // this lane-half's K-pair row
            v2f a0 = Xp[kpl * LDN + m0      + row];   // A: rows m0..m0+15
            v2f a1 = Xp[kpl * LDN + m0 + 16 + row];   // A: rows m0+16..m0+31
            v2f b0 = Yp[kpl * LDN + n0      + row];   // B: cols n0..n0+15
            v2f b1 = Yp[kpl * LDN + n0 + 16 + row];   // B: cols n0+16..n0+31

            acc[0][0] = __builtin_amdgcn_wmma_f32_16x16x4_f32(false, a0, false, b0, (short)0, acc[0][0], false, false);
            acc[0][1] = __builtin_amdgcn_wmma_f32_16x16x4_f32(false, a0, false, b1, (short)0, acc[0][1], false, false);
            acc[1][0] = __builtin_amdgcn_wmma_f32_16x16x4_f32(false, a1, false, b0, (short)0, acc[1][0], false, false);
            acc[1][1] = __builtin_amdgcn_wmma_f32_16x16x4_f32(false, a1, false, b1, (short)0, acc[1][1], false, false);
        }

        // ---- stage tile kt+1 into the other buffer ----
        if (have_next) {
            float* xb = (float*)&XsP[cur ^ 1][0];
            float* yb = (float*)&YsP[cur ^ 1][0];
            #pragma unroll
            for (int j = 0; j < 4; ++j) {
                const int c   = cbase + 4 * j;
                const int idx = ((c >> 1) * LDN + nloc) * 2 + (c & 1);
                xb[idx] = xr[j];
                yb[idx] = yr[j];
                sxx += xr[j] * xr[j];
                syy += yr[j] * yr[j];
            }
            __syncthreads();
        }
    }

    const float inv_c = 1.0f / (float)C_DIM;

    // Write xy: C/D layout (16x16 f32): lane L, VGPR r -> M = r + 8*(L>>4), N = L&15
    #pragma unroll
    for (int i = 0; i < 2; ++i) {
        #pragma unroll
        for (int j = 0; j < 2; ++j) {
            const int mm = m0 + 16 * i + kh * 8;
            const int nn = n0 + 16 * j + (lane & 15);
            #pragma unroll
            for (int r = 0; r < 8; ++r) {
                out_xy[(size_t)((mm + r) * N2_ + nn) * WH + wh] = acc[i][j][r] * inv_c;
            }
        }
    }

    // Reduce the 4 per-n partials for xx / yy and store
    __syncthreads();
    red[      (t >> 7) * 128 + nloc] = sxx;
    red[512 + (t >> 7) * 128 + nloc] = syy;
    __syncthreads();
    if (t < 128) {
        const float vx = red[t] + red[128 + t] + red[256 + t] + red[384 + t];
        const float vy = red[512 + t] + red[640 + t] + red[768 + t] + red[896 + t];
        out_xx[(size_t)t * WH + wh] = vx * inv_c;
        out_yy[(size_t)t * WH + wh] = vy * inv_c;
    }
}

extern "C" void kernel_launch(void* const* d_in, const int* in_sizes, int n_in,
                              void* d_out, int out_size, void* d_ws, size_t ws_size,
                              hipStream_t stream) {
    (void)in_sizes; (void)n_in; (void)out_size; (void)d_ws; (void)ws_size;
    const float* x = (const float*)d_in[0];
    const float* y = (const float*)d_in[1];
    float* out    = (float*)d_out;
    float* out_xy = out;                                  // 128*128*256 = 4,194,304
    float* out_xx = out + (size_t)N1_ * N2_ * WH;         // + 32,768
    float* out_yy = out_xx + (size_t)N1_ * WH;            // + 32,768
    nngp_convkp_kernel<<<WH, 512, 0, stream>>>(x, y, out_xy, out_xx, out_yy);
}